// FilmLayerNetwork_73985106641462
// MI455X (gfx1250) — compile-verified
//
#include <hip/hip_runtime.h>
#include <math.h>

#define NMAPS 512
#define ZG    512
#define HID    96
#define SEQL    8
#define NPROC   8

typedef __attribute__((ext_vector_type(2))) float v2f;
typedef __attribute__((ext_vector_type(8))) float v8f;

// ---------------------------------------------------------------------------
// Kernel 1: alpha = sigmoid(x@Wa+ba), q = relu(x@Wqr+bqr), de = (De*gate).sum(-1)
// ---------------------------------------------------------------------------
__global__ void __launch_bounds__(256) prep_kernel(
    const float* __restrict__ gate, const float* __restrict__ x,
    const float* __restrict__ Wa,   const float* __restrict__ ba,
    const float* __restrict__ Wqr,  const float* __restrict__ bqr,
    const float* __restrict__ De,   float* __restrict__ ws) {
  __shared__ float sx[ZG];
  for (int i = threadIdx.x; i < ZG; i += blockDim.x) sx[i] = x[i];
  __syncthreads();

  int t = blockIdx.x * blockDim.x + threadIdx.x;
  float* alpha = ws;            // [512]
  float* qv    = ws + 512;      // [96]
  float* de    = ws + 608;      // [4096]

  if (t < 512) {
    float acc = ba[t];
    #pragma unroll 4
    for (int k = 0; k < ZG; ++k) acc += sx[k] * Wa[k * 512 + t];
    alpha[t] = 1.0f / (1.0f + expf(-acc));
  } else if (t < 608) {
    int j = t - 512;
    float acc = bqr[j];
    #pragma unroll 4
    for (int k = 0; k < ZG; ++k) acc += sx[k] * Wqr[k * 96 + j];
    qv[j] = fmaxf(acc, 0.0f);
  } else if (t < 608 + 4096) {
    int i = t - 608;
    float acc = 0.0f;
    #pragma unroll
    for (int s = 0; s < SEQL; ++s) acc += De[i * 8 + s] * gate[s];
    de[i] = acc;
  }
}

// ---------------------------------------------------------------------------
// WMMA helper: sOut(8x96, ld=96) = sA(8 x kdim, LDS, ld=lda) @ B(kdim x 96, global)
// M padded 8->16; lanes with r>=8 load a duplicate valid row (r&7) — rows 8..15
// of D depend only on rows 8..15 of A, and we never store them, so no zero-fill
// or predication is needed. A fragment = one aligned ds_load_b64 per lane.
// ---------------------------------------------------------------------------
__device__ inline void mm8_wmma(const float* sA, int kdim, int lda,
                                const float* __restrict__ B,
                                float* sOut, int lane) {
  const int r     = lane & 15;           // N column within tile / A row id
  const int ra    = lane & 7;            // duplicated A row for padding lanes
  const int khalf = (lane & 16) ? 2 : 0; // K-half split per fragment layout
  for (int nt = 0; nt < 6; ++nt) {
    const int n0 = nt * 16;
    const float* Ap = sA + ra * lda + khalf;       // 8B-aligned (lda, khalf even)
    const float* Bp = B + khalf * 96 + n0 + r;
    v8f acc = {};
    #pragma unroll 4
    for (int k0 = 0; k0 < kdim; k0 += 4) {
      v2f a = *(const v2f*)Ap;                     // ds_load_b64, unconditional
      v2f b;
      b[0] = Bp[0];
      b[1] = Bp[96];
      if ((k0 & 15) == 0) __builtin_prefetch(Bp + 16 * 96, 0, 1);
      acc = __builtin_amdgcn_wmma_f32_16x16x4_f32(false, a, false, b,
                                                  (short)0, acc, false, false);
      Ap += 4;
      Bp += 4 * 96;
    }
    // C layout: rows 0-7 live in lanes 0-15, VGPRs 0-7 (rows 8-15 discarded)
    if (lane < 16) {
      #pragma unroll
      for (int v = 0; v < 8; ++v) sOut[v * 96 + n0 + lane] = acc[v];
    }
  }
}

// ---------------------------------------------------------------------------
// Kernel 2: one wave32 workgroup per processor (8 blocks).
// ---------------------------------------------------------------------------
__global__ void __launch_bounds__(32) smab_kernel(
    const float* __restrict__ P,
    const float* __restrict__ Wq1, const float* __restrict__ Wk1,
    const float* __restrict__ Wv1, const float* __restrict__ fc1,
    const float* __restrict__ Wq2, const float* __restrict__ Wk2,
    const float* __restrict__ Wv2, const float* __restrict__ fc2,
    const float* __restrict__ Wo,
    const float* __restrict__ ws_q, float* __restrict__ trans) {
  const int p = blockIdx.x;
  const int lane = threadIdx.x;

  const int LDK = 516;                       // padded (even) ld for the 8x512 K-set
  __shared__ __align__(16) float sK[8 * LDK];
  __shared__ __align__(16) float sQk[8 * 96], sKk[8 * 96], sVk[8 * 96];
  __shared__ __align__(16) float sH[8 * 96], sH2[8 * 96], sT[8 * 96];
  __shared__ __align__(16) float sK2[8 * 96], sV2[8 * 96];
  __shared__ float sq[96], sQq[96], sO[96], sT2[96];
  __shared__ float sS[3 * 8 * 8];
  __shared__ float sS2[16 * 8];

  // K_all[p][s][m] = P[p][m][s]   (P: (8,512,8) flat)
  const float* Pp = P + p * (512 * 8);
  for (int e = lane; e < 4096; e += 32) {
    int m = e >> 3, s = e & 7;
    sK[s * LDK + m] = Pp[e];
  }
  for (int e = lane; e < 96; e += 32) sq[e] = ws_q[e];
  __syncthreads();

  const float* wq1 = Wq1 + p * (512 * 96);
  const float* wk1 = Wk1 + p * (512 * 96);
  const float* wv1 = Wv1 + p * (512 * 96);
  const float* f1  = fc1 + p * (96 * 96);
  const float* wq2 = Wq2 + p * (96 * 96);
  const float* wk2 = Wk2 + p * (96 * 96);
  const float* wv2 = Wv2 + p * (96 * 96);
  const float* f2  = fc2 + p * (96 * 96);
  const float* wo  = Wo  + p * (96 * 512);

  // ---- stage 1 projections: Qk/Kk/Vk = K(8x512) @ W(512x96) --------------
  mm8_wmma(sK, 512, LDK, wq1, sQk, lane);
  mm8_wmma(sK, 512, LDK, wk1, sKk, lane);
  mm8_wmma(sK, 512, LDK, wv1, sVk, lane);
  __syncthreads();

  const float scale = rsqrtf(96.0f);

  // ---- stage 1 MHA: 3 heads, d=32, 8x8 scores per head -------------------
  for (int t = 0; t < 6; ++t) {
    int idx = lane + 32 * t;                 // 0..191
    int h = idx >> 6, rem = idx & 63;
    int i = rem >> 3, j = rem & 7;
    float s = 0.0f;
    #pragma unroll 8
    for (int d = 0; d < 32; ++d)
      s += sQk[i * 96 + h * 32 + d] * sKk[j * 96 + h * 32 + d];
    sS[idx] = s * scale;
  }
  __syncthreads();
  if (lane < 24) {                           // softmax over 8, rows (h,i)
    float* row = sS + lane * 8;
    float m = row[0];
    #pragma unroll
    for (int j = 1; j < 8; ++j) m = fmaxf(m, row[j]);
    float e[8], sum = 0.0f;
    #pragma unroll
    for (int j = 0; j < 8; ++j) { e[j] = expf(row[j] - m); sum += e[j]; }
    float inv = 1.0f / sum;
    #pragma unroll
    for (int j = 0; j < 8; ++j) row[j] = e[j] * inv;
  }
  __syncthreads();
  for (int t = 0; t < 24; ++t) {             // H = Qk + A@V
    int idx = lane + 32 * t;                 // 0..767
    int i = idx / 96, n = idx % 96;
    int h = n / 32;
    float o = 0.0f;
    #pragma unroll
    for (int j = 0; j < 8; ++j) o += sS[h * 64 + i * 8 + j] * sVk[j * 96 + n];
    sH[idx] = sQk[idx] + o;
  }
  __syncthreads();

  // ---- H = H + relu(H @ fc1) --------------------------------------------
  mm8_wmma(sH, 96, 96, f1, sT, lane);
  __syncthreads();
  for (int t = 0; t < 24; ++t) {
    int idx = lane + 32 * t;
    sH2[idx] = sH[idx] + fmaxf(sT[idx], 0.0f);
  }
  __syncthreads();

  // ---- stage 2: Qq = q @ wq2 (1x96); K2/V2 = H @ wk2/wv2 -----------------
  for (int t = 0; t < 3; ++t) {
    int n = lane + 32 * t;
    float acc = 0.0f;
    #pragma unroll 4
    for (int k = 0; k < 96; ++k) acc += sq[k] * wq2[k * 96 + n];
    sQq[n] = acc;
  }
  mm8_wmma(sH2, 96, 96, wk2, sK2, lane);
  mm8_wmma(sH2, 96, 96, wv2, sV2, lane);
  __syncthreads();

  // ---- stage 2 MHA: 16 heads, d=6, 1 query ------------------------------
  for (int t = 0; t < 4; ++t) {
    int idx = lane + 32 * t;                 // 0..127
    int h = idx >> 3, j = idx & 7;
    float s = 0.0f;
    #pragma unroll
    for (int d = 0; d < 6; ++d) s += sQq[h * 6 + d] * sK2[j * 96 + h * 6 + d];
    sS2[idx] = s * scale;
  }
  __syncthreads();
  if (lane < 16) {
    float* row = sS2 + lane * 8;
    float m = row[0];
    #pragma unroll
    for (int j = 1; j < 8; ++j) m = fmaxf(m, row[j]);
    float e[8], sum = 0.0f;
    #pragma unroll
    for (int j = 0; j < 8; ++j) { e[j] = expf(row[j] - m); sum += e[j]; }
    float inv = 1.0f / sum;
    #pragma unroll
    for (int j = 0; j < 8; ++j) row[j] = e[j] * inv;
  }
  __syncthreads();
  for (int t = 0; t < 3; ++t) {              // O = Qq + A@V2
    int n = lane + 32 * t;
    int h = n / 6;
    float att = 0.0f;
    #pragma unroll
    for (int j = 0; j < 8; ++j) att += sS2[h * 8 + j] * sV2[j * 96 + n];
    sO[n] = sQq[n] + att;
  }
  __syncthreads();
  for (int t = 0; t < 3; ++t) {              // O = O + relu(O @ fc2)
    int n = lane + 32 * t;
    float acc = 0.0f;
    #pragma unroll 4
    for (int k = 0; k < 96; ++k) acc += sO[k] * f2[k * 96 + n];
    sT2[n] = acc;
  }
  __syncthreads();
  for (int t = 0; t < 3; ++t) {
    int n = lane + 32 * t;
    sO[n] = sO[n] + fmaxf(sT2[n], 0.0f);
  }
  __syncthreads();

  // ---- out = O @ Wo : (1x96)@(96x512) ------------------------------------
  float* outp = trans + p * 512;
  for (int t = 0; t < 16; ++t) {
    int n = lane + 32 * t;
    float acc = 0.0f;
    #pragma unroll 4
    for (int k = 0; k < 96; ++k) acc += sO[k] * wo[k * 512 + n];
    outp[n] = acc;
  }
}

// ---------------------------------------------------------------------------
// Kernel 3: FiLM mix
// ---------------------------------------------------------------------------
__global__ void __launch_bounds__(256) final_kernel(
    const float* __restrict__ regs, const float* __restrict__ ws,
    float* __restrict__ out) {
  int i = blockIdx.x * blockDim.x + threadIdx.x;
  if (i >= 4096) return;
  const float* alpha = ws;
  const float* de    = ws + 608;
  const float* trans = ws + 4704;
  int m = i & 511;
  int c = (i >> 9) & 3;
  float a = alpha[m];
  float mix = a * trans[i] + (1.0f - a) * de[i];
  out[i] = mix * regs[i] + (((c & 1) == 0) ? 1.0f : 0.0f);
}

// ---------------------------------------------------------------------------
extern "C" void kernel_launch(void* const* d_in, const int* in_sizes, int n_in,
                              void* d_out, int out_size, void* d_ws, size_t ws_size,
                              hipStream_t stream) {
  const float* gate = (const float*)d_in[0];
  const float* x    = (const float*)d_in[1];
  const float* Wa   = (const float*)d_in[2];
  const float* ba   = (const float*)d_in[3];
  const float* Wqr  = (const float*)d_in[4];
  const float* bqr  = (const float*)d_in[5];
  const float* P    = (const float*)d_in[6];
  const float* De   = (const float*)d_in[7];
  const float* regs = (const float*)d_in[8];
  const float* Wq1  = (const float*)d_in[9];
  const float* Wk1  = (const float*)d_in[10];
  const float* Wv1  = (const float*)d_in[11];
  const float* fc1  = (const float*)d_in[12];
  const float* Wq2  = (const float*)d_in[13];
  const float* Wk2  = (const float*)d_in[14];
  const float* Wv2  = (const float*)d_in[15];
  const float* fc2  = (const float*)d_in[16];
  const float* Wo   = (const float*)d_in[17];
  float* ws  = (float*)d_ws;
  float* out = (float*)d_out;

  prep_kernel<<<19, 256, 0, stream>>>(gate, x, Wa, ba, Wqr, bqr, De, ws);
  smab_kernel<<<NPROC, 32, 0, stream>>>(P, Wq1, Wk1, Wv1, fc1, Wq2, Wk2, Wv2,
                                        fc2, Wo, ws + 512, ws + 4704);
  final_kernel<<<16, 256, 0, stream>>>(regs, ws, out);
}